// VoxelGrid_17514876634252
// MI455X (gfx1250) — compile-verified
//
#include <hip/hip_runtime.h>
#include <math.h>

typedef float v2f __attribute__((ext_vector_type(2)));
typedef float v8f __attribute__((ext_vector_type(8)));

#define GRID_R 160

// One wave (32 lanes) processes 16 points: point p handled by lanes (p, p+16).
// Lane half h = dx side of the trilerp cell. SH deg-2 basis computed via two
// fp32 V_WMMA_F32_16X16X4_F32 ops: basis = P(9x8) * mono(8) + C(consts).
__global__ __launch_bounds__(256) void voxel_sh_kernel(
    const float* __restrict__ points,
    const float* __restrict__ dirs,
    const float* __restrict__ density,
    const float* __restrict__ sh,
    float* __restrict__ out,
    int npts)
{
    const int lane = threadIdx.x & 31;
    const int sub  = lane & 15;      // point index within the wave's group of 16
    const int h    = lane >> 4;      // 0 -> dx=0 corners, 1 -> dx=1 corners
    const int wave = threadIdx.x >> 5;
    const int group = blockIdx.x * (blockDim.x >> 5) + wave;
    int pt = group * 16 + sub;
    pt = pt < npts ? pt : npts - 1;  // clamp: keeps EXEC all-ones for WMMA

    // ---- load point & direction (both lanes of a pair load the same point)
    const float px = points[pt * 3 + 0];
    const float py = points[pt * 3 + 1];
    const float pz = points[pt * 3 + 2];
    const float dxr = dirs[pt * 3 + 0];
    const float dyr = dirs[pt * 3 + 1];
    const float dzr = dirs[pt * 3 + 2];

    // ---- voxel-space coords, align_corners=True: p = (pt+1)*0.5*(R-1)
    const float cx = ((px + 1.0f) * 0.5f) * (float)(GRID_R - 1);
    const float cy = ((py + 1.0f) * 0.5f) * (float)(GRID_R - 1);
    const float cz = ((pz + 1.0f) * 0.5f) * (float)(GRID_R - 1);
    const float flx = floorf(cx), fly = floorf(cy), flz = floorf(cz);
    const float fx = cx - flx, fy = cy - fly, fz = cz - flz;
    const int lox = min(max((int)flx, 0), GRID_R - 1);
    const int loy = min(max((int)fly, 0), GRID_R - 1);
    const int loz = min(max((int)flz, 0), GRID_R - 1);
    const int hix = min(max((int)ceilf(cx), 0), GRID_R - 1);
    const int hiy = min(max((int)ceilf(cy), 0), GRID_R - 1);
    const int hiz = min(max((int)ceilf(cz), 0), GRID_R - 1);

    // ---- direction normalization (zero dirs -> +z, as in reference)
    const float n2 = dxr * dxr + dyr * dyr + dzr * dzr;
    float X, Y, Z;
    if (n2 < 1e-8f) { X = 0.0f; Y = 0.0f; Z = 1.0f; }
    else { const float inv = 1.0f / sqrtf(n2); X = dxr * inv; Y = dyr * inv; Z = dzr * inv; }

    // ---- SH basis via fp32 WMMA: D = P * Mono + C, two K=4 slabs.
    // mono K-order: [y, z, x, xy | yz, z^2, xz, x^2-y^2]
    // A 16x4 layout: lanes 0-15 -> (M=lane, K=0 in v0, K=1 in v1);
    //                lanes 16-31 -> (M=lane-16, K=2 in v0, K=3 in v1)
    // B 4x16 layout: lanes 0-15 -> (K=0 in v0, K=1 in v1, N=lane);
    //                lanes 16-31 -> (K=2 in v0, K=3 in v1, N=lane-16)
    // C/D: lane n vgpr r = [M=r, N=n] (lanes 16-31: M=8+r)
    v8f cmat = {};
    v2f a1 = {0.0f, 0.0f}, a2 = {0.0f, 0.0f}, b1, b2;
    if (h == 0) {
        cmat[0] = 0.282095f;       // b0 constant
        cmat[6] = -0.315392f;      // b6 constant term
        b1[0] = Y;      b1[1] = Z;
        b2[0] = Y * Z;  b2[1] = Z * Z;
        if (sub == 1) a1[0] = 0.488603f;   // P[1,0]: b1 = c*y
        if (sub == 2) a1[1] = 0.488603f;   // P[2,1]: b2 = c*z
        if (sub == 5) a2[0] = 1.092548f;   // P[5,4]: b5 = c*yz
        if (sub == 6) a2[1] = 0.946176f;   // P[6,5]: b6 = c*z^2 - 0.315392
    } else {
        b1[0] = X;      b1[1] = X * Y;
        b2[0] = X * Z;  b2[1] = X * X - Y * Y;
        if (sub == 3) a1[0] = 0.488603f;   // P[3,2]: b3 = c*x
        if (sub == 4) a1[1] = 1.092548f;   // P[4,3]: b4 = c*xy
        if (sub == 7) a2[0] = 1.092548f;   // P[7,6]: b7 = c*xz
        if (sub == 8) a2[1] = 0.546274f;   // P[8,7]: b8 = c*(x^2-y^2)
    }
    v8f dmat = __builtin_amdgcn_wmma_f32_16x16x4_f32(
        false, a1, false, b1, (short)0, cmat, false, false);
    dmat = __builtin_amdgcn_wmma_f32_16x16x4_f32(
        false, a2, false, b2, (short)0, dmat, false, false);

    // ---- broadcast full basis[0..8] to both lanes of the pair
    float basis[9];
    {
        float oth0 = __shfl_xor(dmat[0], 16, 32);
        basis[0] = (h == 0) ? dmat[0] : oth0;   // lanes>=16 row8 sits in dmat[0]
        basis[8] = (h == 0) ? oth0 : dmat[0];
    }
#pragma unroll
    for (int j = 1; j < 8; ++j) {
        float oth = __shfl_xor(dmat[j], 16, 32);
        basis[j] = (h == 0) ? dmat[j] : oth;
    }

    // ---- trilerp gathers: this lane handles its dx-half (4 corners)
    const int   ix = h ? hix : lox;
    const float wx = h ? fx : (1.0f - fx);

    float shAcc[27];
#pragma unroll
    for (int k = 0; k < 27; ++k) shAcc[k] = 0.0f;
    float sig = 0.0f;

#pragma unroll
    for (int dy = 0; dy < 2; ++dy) {
        const int   iy  = dy ? hiy : loy;
        const float wxy = wx * (dy ? fy : (1.0f - fy));
#pragma unroll
        for (int dz = 0; dz < 2; ++dz) {
            const int   iz = dz ? hiz : loz;
            const float w  = wxy * (dz ? fz : (1.0f - fz));
            const int idx = (ix * GRID_R + iy) * GRID_R + iz;
            sig = fmaf(w, density[idx], sig);
            const float* sp = sh + (long)idx * 27;   // 27 contiguous floats/voxel
#pragma unroll
            for (int k = 0; k < 27; ++k)             // streaming, ~no reuse -> NT
                shAcc[k] = fmaf(w, __builtin_nontemporal_load(sp + k), shAcc[k]);
        }
    }

    // ---- per-lane partial SH dot, then combine the pair
    float t0 = 0.0f, t1 = 0.0f, t2 = 0.0f;
#pragma unroll
    for (int k = 0; k < 9; ++k) {
        t0 = fmaf(shAcc[k],      basis[k], t0);
        t1 = fmaf(shAcc[9 + k],  basis[k], t1);
        t2 = fmaf(shAcc[18 + k], basis[k], t2);
    }
    t0  += __shfl_xor(t0, 16, 32);
    t1  += __shfl_xor(t1, 16, 32);
    t2  += __shfl_xor(t2, 16, 32);
    sig += __shfl_xor(sig, 16, 32);

    if (h == 0) {
        float4 o;
        o.x = sig;
        o.y = 1.0f / (1.0f + expf(-t0));
        o.z = 1.0f / (1.0f + expf(-t1));
        o.w = 1.0f / (1.0f + expf(-t2));
        reinterpret_cast<float4*>(out)[pt] = o;   // 16B-aligned coalesced store
    }
}

extern "C" void kernel_launch(void* const* d_in, const int* in_sizes, int n_in,
                              void* d_out, int out_size, void* d_ws, size_t ws_size,
                              hipStream_t stream) {
    const float* points  = (const float*)d_in[0];
    const float* dirs    = (const float*)d_in[1];
    const float* density = (const float*)d_in[2];
    const float* sh      = (const float*)d_in[3];
    float* out = (float*)d_out;

    const int npts   = in_sizes[0] / 3;            // [N,3] points
    const int groups = (npts + 15) / 16;           // 16 points per wave
    const long threads = (long)groups * 32;
    const int block = 256;                          // 8 wave32s per block
    const int grid  = (int)((threads + block - 1) / block);
    voxel_sh_kernel<<<grid, block, 0, stream>>>(points, dirs, density, sh, out, npts);
}